// GCNLayer_18442589569934
// MI455X (gfx1250) — compile-verified
//
#include <hip/hip_runtime.h>
#include <hip/hip_bf16.h>

typedef float v2f __attribute__((ext_vector_type(2)));
typedef float v8f __attribute__((ext_vector_type(8)));

#define DIM 128   // IN_DIM == OUT_DIM == 128

// ---------------- degree / normalization ----------------

__global__ void gcn_init_deg(unsigned* __restrict__ deg, int n) {
    int i = blockIdx.x * blockDim.x + threadIdx.x;
    if (i < n) deg[i] = 0u;
}

__global__ void gcn_count_deg(const int* __restrict__ ei, unsigned* __restrict__ deg,
                              int n_edges) {
    int e = blockIdx.x * blockDim.x + threadIdx.x;
    if (e < n_edges) atomicAdd(&deg[ei[n_edges + e]], 1u);   // col index
}

__global__ void gcn_dinv(const unsigned* __restrict__ deg, float* __restrict__ dinv, int n) {
    int i = blockIdx.x * blockDim.x + threadIdx.x;
    if (i < n) dinv[i] = rsqrtf((float)deg[i] + 1.0f);       // +1 = self loop
}

// ---------------- h = x @ W^T via V_WMMA_F32_16X16X4_F32 ----------------
// One block = 16 node rows. 8 waves, wave w owns output columns [16w, 16w+16).
// x row-tile (16x128 = 8KB) staged in LDS, W (64KB) stays hot in L2.

__global__ __launch_bounds__(256) void gcn_gemm_wmma(const float* __restrict__ x,
                                                     const float* __restrict__ W,
                                                     float* __restrict__ h,
                                                     int n_nodes) {
    __shared__ float xs[16 * DIM];   // 8 KB

    const int tid     = threadIdx.x;
    const int wave    = tid >> 5;          // 0..7 -> column tile
    const int lane    = tid & 31;
    const int rowbase = blockIdx.x * 16;

    // Cooperative load of the 16x128 x tile into LDS (512 float4, 2 per thread).
    {
        const float4* xv = (const float4*)x;
        float4*       sv = (float4*)xs;
        #pragma unroll
        for (int it = 0; it < 2; ++it) {
            int fi  = tid + it * 256;          // float4 index in tile [0,512)
            int row = rowbase + (fi >> 5);     // 32 float4 per row
            float4 v = make_float4(0.f, 0.f, 0.f, 0.f);
            if (row < n_nodes) v = xv[(size_t)row * 32 + (fi & 31)];
            sv[fi] = v;
        }
    }
    __syncthreads();

    const int half    = lane >> 4;     // 0: K+0/K+1 lanes, 1: K+2/K+3 lanes
    const int l15     = lane & 15;
    const int colbase = wave * 16;

    v8f acc = {};
    #pragma unroll
    for (int k = 0; k < DIM; k += 4) {
        const int ka = k + 2 * half;
        // A: 16x4 fp32 tile of x rows (from LDS)
        v2f a;
        a.x = xs[l15 * DIM + ka];
        a.y = xs[l15 * DIM + ka + 1];
        // B: 4x16 fp32 tile of W^T  => B[k][n] = W[colbase+n][k]
        const float* wrow = W + (size_t)(colbase + l15) * DIM + ka;
        v2f b;
        b.x = wrow[0];
        b.y = wrow[1];
        acc = __builtin_amdgcn_wmma_f32_16x16x4_f32(
            /*neg_a=*/false, a, /*neg_b=*/false, b,
            /*c_mod=*/(short)0, acc, /*reuse_a=*/false, /*reuse_b=*/false);
    }

    // C/D layout: VGPR v -> (M=v, N=lane) for lanes 0-15, (M=v+8, N=lane-16) for 16-31
    #pragma unroll
    for (int v = 0; v < 8; ++v) {
        int row = rowbase + v + 8 * half;
        int col = colbase + l15;
        if (row < n_nodes) h[(size_t)row * DIM + col] = acc[v];
    }
}

// ---------------- out = self-loop term, then scatter-add edges ----------------

__global__ void gcn_init_out(const float* __restrict__ h, const float* __restrict__ dinv,
                             float* __restrict__ out, int n_nodes) {
    int i = blockIdx.x * blockDim.x + threadIdx.x;    // float4 index
    int total = n_nodes * (DIM / 4);
    if (i >= total) return;
    int node = i >> 5;                                // 32 float4 per row
    float w = dinv[node] * dinv[node];
    float4 hv = ((const float4*)h)[i];
    float4 o  = make_float4(w * hv.x, w * hv.y, w * hv.z, w * hv.w);
    ((float4*)out)[i] = o;
}

// One wave per edge; lane owns 4 feature dims (float4 gather + 4 f32 atomics).
__global__ __launch_bounds__(256) void gcn_edge_scatter(const int* __restrict__ ei,
                                                        const float* __restrict__ h,
                                                        const float* __restrict__ dinv,
                                                        float* __restrict__ out,
                                                        int n_edges) {
    const int lane   = threadIdx.x & 31;
    const int wid    = (blockIdx.x * (blockDim.x >> 5)) + (threadIdx.x >> 5);
    const int nwaves = gridDim.x * (blockDim.x >> 5);
    for (int e = wid; e < n_edges; e += nwaves) {
        int r = ei[e];
        int c = ei[n_edges + e];
        float w = dinv[r] * dinv[c];
        float4 hv = ((const float4*)(h + (size_t)c * DIM))[lane];
        float* o = out + (size_t)r * DIM + lane * 4;
        atomicAdd(o + 0, w * hv.x);
        atomicAdd(o + 1, w * hv.y);
        atomicAdd(o + 2, w * hv.z);
        atomicAdd(o + 3, w * hv.w);
    }
}

__global__ void gcn_relu(float* __restrict__ out, int total4) {
    int i = blockIdx.x * blockDim.x + threadIdx.x;
    if (i >= total4) return;
    float4 v = ((float4*)out)[i];
    v.x = fmaxf(v.x, 0.f); v.y = fmaxf(v.y, 0.f);
    v.z = fmaxf(v.z, 0.f); v.w = fmaxf(v.w, 0.f);
    ((float4*)out)[i] = v;
}

// ---------------- launcher ----------------

extern "C" void kernel_launch(void* const* d_in, const int* in_sizes, int n_in,
                              void* d_out, int out_size, void* d_ws, size_t ws_size,
                              hipStream_t stream) {
    const float* x  = (const float*)d_in[0];
    const float* W  = (const float*)d_in[1];
    const int*   ei = (const int*)d_in[2];
    float*       out = (float*)d_out;

    const int n_nodes = in_sizes[0] / DIM;
    const int n_edges = in_sizes[2] / 2;

    // Workspace layout: h [n*128 f32] | dinv [n f32] | deg [n u32]
    float*    h    = (float*)d_ws;
    float*    dinv = h + (size_t)n_nodes * DIM;
    unsigned* deg  = (unsigned*)(dinv + n_nodes);

    const int t = 256;

    gcn_init_deg<<<(n_nodes + t - 1) / t, t, 0, stream>>>(deg, n_nodes);
    gcn_count_deg<<<(n_edges + t - 1) / t, t, 0, stream>>>(ei, deg, n_edges);
    gcn_dinv<<<(n_nodes + t - 1) / t, t, 0, stream>>>(deg, dinv, n_nodes);

    gcn_gemm_wmma<<<(n_nodes + 15) / 16, 256, 0, stream>>>(x, W, h, n_nodes);

    const int total4 = n_nodes * (DIM / 4);
    gcn_init_out<<<(total4 + t - 1) / t, t, 0, stream>>>(h, dinv, out, n_nodes);

    gcn_edge_scatter<<<2048, 256, 0, stream>>>(ei, h, dinv, out, n_edges);

    gcn_relu<<<(total4 + t - 1) / t, t, 0, stream>>>(out, total4);
}